// GraphSAGEEncoder_61220463837359
// MI455X (gfx1250) — compile-verified
//
#include <hip/hip_runtime.h>

typedef float v2f __attribute__((ext_vector_type(2)));
typedef float v8f __attribute__((ext_vector_type(8)));

#define DIM 64
#define WPAD 80   // v2f stride per K-pair row; 80*2 DWORDs = 32 mod 64 banks -> no conflicts

// ---------------------------------------------------------------------------
// zero a float buffer (grid-stride)
__global__ void zero_f32(float* __restrict__ p, long long n) {
    long long i = (long long)blockIdx.x * blockDim.x + threadIdx.x;
    long long stride = (long long)gridDim.x * blockDim.x;
    for (; i < n; i += stride) p[i] = 0.0f;
}

// ---------------------------------------------------------------------------
// in-degree as float via atomics
__global__ void degree_kernel(const int* __restrict__ dst, float* __restrict__ deg,
                              int n_edges) {
    int e = blockIdx.x * blockDim.x + threadIdx.x;
    if (e < n_edges) atomicAdd(&deg[dst[e]], 1.0f);
}

__global__ void invdeg_kernel(float* __restrict__ deg, int n) {
    int i = blockIdx.x * blockDim.x + threadIdx.x;
    if (i < n) deg[i] = 1.0f / fmaxf(deg[i], 1.0f);
}

// ---------------------------------------------------------------------------
// scatter-add: agg[dst[e]][d] += feat[src[e]][d]
// 64 consecutive threads per edge -> each wave32 covers half an edge:
// scalar src/dst per wave, contiguous 128B gather + contiguous atomic region.
// Node features (25.6 MB) are L2-resident -> atomics resolve in L2.
__global__ void scatter_add_kernel(const float* __restrict__ feat,
                                   const int* __restrict__ src,
                                   const int* __restrict__ dst,
                                   float* __restrict__ agg, int n_edges) {
    long long tid = (long long)blockIdx.x * blockDim.x + threadIdx.x;
    int e = (int)(tid >> 6);
    int d = (int)(tid & 63);
    if (e >= n_edges) return;
    int s = src[e];
    int t = dst[e];
    atomicAdd(&agg[(long long)t * DIM + d], feat[(long long)s * DIM + d]);
}

// ---------------------------------------------------------------------------
// Fused SAGEConv node transform using fp32 WMMA (V_WMMA_F32_16X16X4_F32):
//   out[m,:] = (agg[m,:] * invdeg[m]) @ wl + self[m,:] @ wr + bl   (+ReLU)
// One wave32 per 16-node row tile; 4 accumulator tiles of 16x16 cover the
// 64-wide output. Weights are staged once per block into LDS, repacked
// K-pair-major so each lane's B fragment {w[k][n], w[k+1][n]} is a single
// contiguous v2f -> one ds_load_b64 per fragment, bank-conflict-free (WPAD).
template <bool RELU>
__global__ void sage_gemm_kernel(const float* __restrict__ agg,
                                 const float* __restrict__ invdeg,
                                 const float* __restrict__ self_feat,
                                 const float* __restrict__ wl,
                                 const float* __restrict__ bl,
                                 const float* __restrict__ wr,
                                 float* __restrict__ out, int n_nodes) {
    __shared__ v2f lds_wl[32 * WPAD];   // [K-pair p][col n] = {w[2p][n], w[2p+1][n]}
    __shared__ v2f lds_wr[32 * WPAD];

    // Stage + repack both weight matrices (32 K-pairs x 64 cols each).
    // Coalesced b32 reads across columns; 8 iterations for 256 threads.
    for (int i = threadIdx.x; i < 32 * 64; i += blockDim.x) {
        const int p = i >> 6;     // K-pair index
        const int n = i & 63;     // column
        v2f a, b;
        a.x = wl[(2 * p) * DIM + n];
        a.y = wl[(2 * p + 1) * DIM + n];
        b.x = wr[(2 * p) * DIM + n];
        b.y = wr[(2 * p + 1) * DIM + n];
        lds_wl[p * WPAD + n] = a;
        lds_wr[p * WPAD + n] = b;
    }
    __syncthreads();

    const int lane = threadIdx.x & 31;
    const int wave = threadIdx.x >> 5;
    const int tile = blockIdx.x * (blockDim.x >> 5) + wave;
    const int n_tiles = (n_nodes + 15) >> 4;
    if (tile >= n_tiles) return;       // whole-wave uniform exit; EXEC all-1s below

    const int row0 = tile << 4;
    const int r     = lane & 15;       // A: row M within tile; B/D: column N % 16
    const int kh    = (lane >> 4) << 1;// K sub-offset: 0 (lanes 0-15) or 2 (16-31)
    const int pbase = lane >> 4;       // K-pair sub-offset: 0 or 1

    int node = row0 + r;
    if (node > n_nodes - 1) node = n_nodes - 1;   // clamp; results masked on store
    const float id = invdeg[node];
    const float* arow = agg       + (long long)node * DIM;
    const float* xrow = self_feat + (long long)node * DIM;

    v8f acc[4] = {{}, {}, {}, {}};

    // Phase 1: (agg * invdeg) @ wl
#pragma unroll
    for (int k = 0; k < DIM; k += 4) {
        v2f a;
        a.x = arow[k + kh]     * id;   // A[m][k+kh]   (VGPR0)
        a.y = arow[k + kh + 1] * id;   // A[m][k+kh+1] (VGPR1)
        const int prow = (k >> 1) + pbase;
#pragma unroll
        for (int nt = 0; nt < 4; ++nt) {
            v2f b = lds_wl[prow * WPAD + nt * 16 + r];   // ds_load_b64
            acc[nt] = __builtin_amdgcn_wmma_f32_16x16x4_f32(
                false, a, false, b, (short)0, acc[nt], false, false);
        }
    }

    // Phase 2: self @ wr
#pragma unroll
    for (int k = 0; k < DIM; k += 4) {
        v2f a;
        a.x = xrow[k + kh];
        a.y = xrow[k + kh + 1];
        const int prow = (k >> 1) + pbase;
#pragma unroll
        for (int nt = 0; nt < 4; ++nt) {
            v2f b = lds_wr[prow * WPAD + nt * 16 + r];   // ds_load_b64
            acc[nt] = __builtin_amdgcn_wmma_f32_16x16x4_f32(
                false, a, false, b, (short)0, acc[nt], false, false);
        }
    }

    // Epilogue: bias (+ReLU), store. D layout: VGPR vr -> M = vr + 8*(lane>>4),
    // N = lane&15 within the 16-wide column tile.
    const int mbase = (lane >> 4) << 3;   // 0 or 8
#pragma unroll
    for (int nt = 0; nt < 4; ++nt) {
        const int n = nt * 16 + r;
        const float bias = bl[n];
#pragma unroll
        for (int vr = 0; vr < 8; ++vr) {
            const int m = row0 + mbase + vr;
            if (m < n_nodes) {
                float v = acc[nt][vr] + bias;
                if (RELU) v = fmaxf(v, 0.0f);
                out[(long long)m * DIM + n] = v;
            }
        }
    }
}

// ---------------------------------------------------------------------------
extern "C" void kernel_launch(void* const* d_in, const int* in_sizes, int n_in,
                              void* d_out, int out_size, void* d_ws, size_t ws_size,
                              hipStream_t stream) {
    const float* x   = (const float*)d_in[0];
    const int*  edge = (const int*) d_in[1];
    const float* w1l = (const float*)d_in[2];
    const float* b1l = (const float*)d_in[3];
    const float* w1r = (const float*)d_in[4];
    const float* w2l = (const float*)d_in[5];
    const float* b2l = (const float*)d_in[6];
    const float* w2r = (const float*)d_in[7];
    float* out = (float*)d_out;

    const int n_nodes = in_sizes[0] / DIM;
    const int n_edges = in_sizes[1] / 2;
    const int* src = edge;             // edge_index[0]
    const int* dst = edge + n_edges;   // edge_index[1]

    // workspace: invdeg [N] | agg [N*64] (reused by both layers) | h [N*64]
    float* ws      = (float*)d_ws;
    float* invdeg  = ws;
    float* agg     = ws + n_nodes;
    float* h       = agg + (long long)n_nodes * DIM;

    const long long feat_elems = (long long)n_nodes * DIM;
    const int edge_blocks  = (n_edges + 255) / 256;
    const int node_blocks  = (n_nodes + 255) / 256;
    const int sc_blocks    = (int)(((long long)n_edges * DIM + 255) / 256);
    const int tiles        = (n_nodes + 15) / 16;
    const int gemm_blocks  = (tiles + 7) / 8;   // 8 waves (256 thr) per block

    // --- degree / inverse degree (shared by both layers) ---
    zero_f32<<<1024, 256, 0, stream>>>(invdeg, n_nodes);
    degree_kernel<<<edge_blocks, 256, 0, stream>>>(dst, invdeg, n_edges);
    invdeg_kernel<<<node_blocks, 256, 0, stream>>>(invdeg, n_nodes);

    // --- layer 1: scatter(x) -> WMMA transform (+ReLU) -> h ---
    zero_f32<<<2048, 256, 0, stream>>>(agg, feat_elems);
    scatter_add_kernel<<<sc_blocks, 256, 0, stream>>>(x, src, dst, agg, n_edges);
    sage_gemm_kernel<true><<<gemm_blocks, 256, 0, stream>>>(
        agg, invdeg, x, w1l, b1l, w1r, h, n_nodes);

    // --- layer 2: scatter(h) -> WMMA transform -> out ---
    zero_f32<<<2048, 256, 0, stream>>>(agg, feat_elems);
    scatter_add_kernel<<<sc_blocks, 256, 0, stream>>>(h, src, dst, agg, n_edges);
    sage_gemm_kernel<false><<<gemm_blocks, 256, 0, stream>>>(
        agg, invdeg, h, w2l, b2l, w2r, out, n_nodes);
}